// NeuralDictionaryV11_38594576121966
// MI455X (gfx1250) — compile-verified
//
#include <hip/hip_runtime.h>
#include <math.h>

#define N_CAP 1000000
#define D_DIM 64
#define V_DIM 64

typedef float v2f __attribute__((ext_vector_type(2)));
typedef float v8f __attribute__((ext_vector_type(8)));

// ---------------------------------------------------------------------------
// Pass 1: scores[i] = -sum_j |keys[i][j] - query[j]|, plus per-block max.
// Bandwidth-bound: streams 256 MB of keys once. One row per thread, float4
// loads (each lane streams a contiguous 256B row; all bytes used).
// ---------------------------------------------------------------------------
__global__ void score_kernel(const float* __restrict__ query,
                             const float* __restrict__ keys,
                             float* __restrict__ scores,
                             float* __restrict__ pmax) {
    __shared__ float red[256];
    int i = blockIdx.x * 256 + threadIdx.x;
    float s = -INFINITY;
    if (i < N_CAP) {
        const float4* q4 = (const float4*)query;
        const float4* k4 = (const float4*)(keys + (size_t)i * D_DIM);
        float acc = 0.0f;
#pragma unroll
        for (int j = 0; j < D_DIM / 4; ++j) {
            float4 q = q4[j];
            float4 k = k4[j];
            acc += fabsf(k.x - q.x) + fabsf(k.y - q.y) +
                   fabsf(k.z - q.z) + fabsf(k.w - q.w);
        }
        s = -acc;
        scores[i] = s;
    }
    red[threadIdx.x] = s;
    __syncthreads();
    for (int off = 128; off > 0; off >>= 1) {
        if (threadIdx.x < off)
            red[threadIdx.x] = fmaxf(red[threadIdx.x], red[threadIdx.x + off]);
        __syncthreads();
    }
    if (threadIdx.x == 0) pmax[blockIdx.x] = red[0];
}

// ---------------------------------------------------------------------------
// Single-block reductions over per-block partials (fixed order => deterministic)
// ---------------------------------------------------------------------------
__global__ void reduce_max_kernel(const float* __restrict__ in, int n,
                                  float* __restrict__ out) {
    __shared__ float red[1024];
    float m = -INFINITY;
    for (int i = threadIdx.x; i < n; i += 1024) m = fmaxf(m, in[i]);
    red[threadIdx.x] = m;
    __syncthreads();
    for (int off = 512; off > 0; off >>= 1) {
        if (threadIdx.x < off)
            red[threadIdx.x] = fmaxf(red[threadIdx.x], red[threadIdx.x + off]);
        __syncthreads();
    }
    if (threadIdx.x == 0) *out = red[0];
}

__global__ void reduce_sum_kernel(const float* __restrict__ in, int n,
                                  float* __restrict__ out) {
    __shared__ float red[1024];
    float s = 0.0f;
    for (int i = threadIdx.x; i < n; i += 1024) s += in[i];
    red[threadIdx.x] = s;
    __syncthreads();
    for (int off = 512; off > 0; off >>= 1) {
        if (threadIdx.x < off)
            red[threadIdx.x] += red[threadIdx.x + off];
        __syncthreads();
    }
    if (threadIdx.x == 0) *out = red[0];
}

// ---------------------------------------------------------------------------
// Pass 2: e[i] = exp(s[i] - m1) in-place, per-block sum (first softmax numer.)
// ---------------------------------------------------------------------------
__global__ void exp1_kernel(float* __restrict__ data,
                            const float* __restrict__ m1p,
                            float* __restrict__ psum) {
    __shared__ float red[256];
    int i = blockIdx.x * 256 + threadIdx.x;
    float m1 = *m1p;
    float e = 0.0f;
    if (i < N_CAP) {
        e = __expf(data[i] - m1);
        data[i] = e;
    }
    red[threadIdx.x] = e;
    __syncthreads();
    for (int off = 128; off > 0; off >>= 1) {
        if (threadIdx.x < off) red[threadIdx.x] += red[threadIdx.x + off];
        __syncthreads();
    }
    if (threadIdx.x == 0) psum[blockIdx.x] = red[0];
}

// ---------------------------------------------------------------------------
// Pass 3: w[i] = exp(e[i]/Z1) in-place, per-block sum (second softmax numer.;
// p_i = e_i/Z1 is in (0,1] so no max-subtraction needed — no overflow possible)
// ---------------------------------------------------------------------------
__global__ void exp2_kernel(float* __restrict__ data,
                            const float* __restrict__ z1p,
                            float* __restrict__ psum) {
    __shared__ float red[256];
    int i = blockIdx.x * 256 + threadIdx.x;
    float rz = 1.0f / *z1p;
    float w = 0.0f;
    if (i < N_CAP) {
        w = __expf(data[i] * rz);
        data[i] = w;
    }
    red[threadIdx.x] = w;
    __syncthreads();
    for (int off = 128; off > 0; off >>= 1) {
        if (threadIdx.x < off) red[threadIdx.x] += red[threadIdx.x + off];
        __syncthreads();
    }
    if (threadIdx.x == 0) psum[blockIdx.x] = red[0];
}

// ---------------------------------------------------------------------------
// Pass 4 (the big one, streams 256 MB of values): pout[b][n] = sum_i w_i*v[i][n]
// via V_WMMA_F32_16X16X4_F32. Per 4-row chunk:
//   A (16x4 f32, 2 VGPRs): every M row = (w0,w1,w2,w3).
//     Layout: lanes 0-15 hold K=0 (VGPR0) / K=1 (VGPR1); lanes 16-31 hold K=2/3.
//   B (4x16 f32, 2 VGPRs): B[k][n] = values[i0+k][n0+n], same K split, N=lane&15.
//   D: all 16 rows identical; D[0] of lanes 0-15 = partial out[n0 + lane].
// V=64 -> 4 accumulators/wave. Loop bound is wave-uniform, no divergence in
// the WMMA region (EXEC all-1s as required).
// ---------------------------------------------------------------------------
__global__ void wmma_accum_kernel(const float* __restrict__ values,
                                  const float* __restrict__ w,
                                  float* __restrict__ pout) {
    __shared__ float sOut[8][64];
    const int lane = threadIdx.x & 31;
    const int wave = threadIdx.x >> 5;
    const int half = lane >> 4;   // 0: K=0,1   1: K=2,3
    const int li   = lane & 15;   // N within 16-col tile

    v8f acc0 = {}, acc1 = {}, acc2 = {}, acc3 = {};

    const int nChunks = N_CAP / 4;           // 250000, exact
    const int gw      = blockIdx.x * 8 + wave;
    const int stride  = gridDim.x * 8;

    for (int c = gw; c < nChunks; c += stride) {
        const int i0 = c * 4;
        const float4 wv = *(const float4*)(w + i0);   // uniform broadcast, L2-hot

        v2f a;
        a.x = half ? wv.z : wv.x;
        a.y = half ? wv.w : wv.y;

        const float* vrow0 = values + (size_t)(i0 + 2 * half) * V_DIM;
        const float* vrow1 = vrow0 + V_DIM;

        v2f b0; b0.x = vrow0[li];      b0.y = vrow1[li];
        v2f b1; b1.x = vrow0[16 + li]; b1.y = vrow1[16 + li];
        v2f b2; b2.x = vrow0[32 + li]; b2.y = vrow1[32 + li];
        v2f b3; b3.x = vrow0[48 + li]; b3.y = vrow1[48 + li];

        acc0 = __builtin_amdgcn_wmma_f32_16x16x4_f32(false, a, false, b0,
                                                     (short)0, acc0, false, false);
        acc1 = __builtin_amdgcn_wmma_f32_16x16x4_f32(false, a, false, b1,
                                                     (short)0, acc1, false, false);
        acc2 = __builtin_amdgcn_wmma_f32_16x16x4_f32(false, a, false, b2,
                                                     (short)0, acc2, false, false);
        acc3 = __builtin_amdgcn_wmma_f32_16x16x4_f32(false, a, false, b3,
                                                     (short)0, acc3, false, false);
    }

    // One contribution per column: lanes 0-15 carry M=0 row in element 0.
    if (lane < 16) {
        sOut[wave][ 0 + li] = acc0[0];
        sOut[wave][16 + li] = acc1[0];
        sOut[wave][32 + li] = acc2[0];
        sOut[wave][48 + li] = acc3[0];
    }
    __syncthreads();
    if (threadIdx.x < 64) {
        float s = 0.0f;
#pragma unroll
        for (int wv = 0; wv < 8; ++wv) s += sOut[wv][threadIdx.x];
        pout[blockIdx.x * 64 + threadIdx.x] = s;
    }
}

// ---------------------------------------------------------------------------
// Finalize: out[n] = (sum_b pout[b][n]) / Z2   (fixed-order => deterministic)
// ---------------------------------------------------------------------------
__global__ void finalize_kernel(const float* __restrict__ pout, int nb,
                                const float* __restrict__ z2p,
                                float* __restrict__ out) {
    int n = threadIdx.x;  // 64 threads
    float s = 0.0f;
    for (int b = 0; b < nb; ++b) s += pout[(size_t)b * 64 + n];
    out[n] = s / (*z2p);
}

extern "C" void kernel_launch(void* const* d_in, const int* in_sizes, int n_in,
                              void* d_out, int out_size, void* d_ws, size_t ws_size,
                              hipStream_t stream) {
    const float* query  = (const float*)d_in[0];
    const float* keys   = (const float*)d_in[1];
    const float* values = (const float*)d_in[2];
    float* out = (float*)d_out;

    float* ws     = (float*)d_ws;
    float* scores = ws;                        // N floats (reused: s -> e -> w)
    float* pmax   = ws + N_CAP;                // 4096
    float* psum   = ws + N_CAP + 4096;         // 4096
    float* psum2  = ws + N_CAP + 8192;         // 4096
    float* scal   = ws + N_CAP + 12288;        // [0]=m1 [1]=Z1 [2]=Z2
    float* pout   = ws + N_CAP + 12352;        // B7 * 64

    const int B1 = (N_CAP + 255) / 256;        // 3907
    const int B7 = 512;                        // 4096 waves over 250000 chunks

    score_kernel<<<B1, 256, 0, stream>>>(query, keys, scores, pmax);
    reduce_max_kernel<<<1, 1024, 0, stream>>>(pmax, B1, &scal[0]);
    exp1_kernel<<<B1, 256, 0, stream>>>(scores, &scal[0], psum);
    reduce_sum_kernel<<<1, 1024, 0, stream>>>(psum, B1, &scal[1]);
    exp2_kernel<<<B1, 256, 0, stream>>>(scores, &scal[1], psum2);
    reduce_sum_kernel<<<1, 1024, 0, stream>>>(psum2, B1, &scal[2]);
    wmma_accum_kernel<<<B7, 256, 0, stream>>>(values, scores, pout);
    finalize_kernel<<<1, 64, 0, stream>>>(pout, B7, &scal[2], out);
}